// GAT_cat_decoder_12738873000211
// MI455X (gfx1250) — compile-verified
//
#include <hip/hip_runtime.h>
#include <math.h>

#define ATT_SLOPE 0.2f
#define ACT_SLOPE 0.01f

typedef __attribute__((ext_vector_type(2))) float v2f;
typedef __attribute__((ext_vector_type(8))) float v8f;

__device__ __forceinline__ float lrelu(float v, float s) { return v > 0.f ? v : v * s; }

// Float atomic max via integer atomics (monotonic bit trick; works with -inf init).
__device__ __forceinline__ void atomicMaxF(float* addr, float val) {
  if (val >= 0.f)
    atomicMax((int*)addr, __float_as_int(val));
  else
    atomicMin((unsigned int*)addr, (unsigned int)__float_as_int(val));
}

// ---------------------------------------------------------------------------
// D[M x NCOLS] = A[M x K] * B[K x NCOLS], f32, via V_WMMA_F32_16X16X4_F32.
// One wave (32 lanes, EXEC all-1s) per 16x16 output tile; K fully unrolled.
// A 16x4 tile: lanes 0-15 carry K=k,k+1 ; lanes 16-31 carry K=k+2,k+3.
// B 4x16 tile mirrors (rows striped across the two lane halves).
// C/D: VGPR r -> row (half*8 + r), col = lane&15.
// ---------------------------------------------------------------------------
template <int K, int NCOLS>
__global__ void __launch_bounds__(32)
wmma_gemm_f32(const float* __restrict__ A, const float* __restrict__ B,
              float* __restrict__ D, int M) {
  const int m0 = blockIdx.x * 16;
  const int n0 = blockIdx.y * 16;
  const int lane = threadIdx.x;
  const int half = lane >> 4;   // 0 or 1
  const int lid  = lane & 15;
  const int ar = min(m0 + lid, M - 1);  // clamped load row keeps EXEC uniform
  v8f acc = {};
#pragma unroll
  for (int k = 0; k < K; k += 4) {
    const int ka = k + half * 2;
    v2f a, b;
    a.x = A[(size_t)ar * K + ka];
    a.y = A[(size_t)ar * K + ka + 1];
    b.x = B[(size_t)ka * NCOLS + n0 + lid];
    b.y = B[(size_t)(ka + 1) * NCOLS + n0 + lid];
    acc = __builtin_amdgcn_wmma_f32_16x16x4_f32(false, a, false, b, (short)0, acc,
                                                false, false);
  }
  const int rbase = m0 + half * 8;
#pragma unroll
  for (int r = 0; r < 8; ++r) {
    const int row = rbase + r;
    if (row < M) D[(size_t)row * NCOLS + n0 + lid] = acc[r];
  }
}

// Coalesced flat zero-fill.
__global__ void zero_buf(float* __restrict__ p, unsigned n) {
  const unsigned t = blockIdx.x * blockDim.x + threadIdx.x;
  if (t < n) p[t] = 0.f;
}

// Per-node: attention coefficients a_s = <h[n,h,:],att_src[h]>, a_d likewise;
// init segment-max to -inf and denom to 0.
__global__ void node_prep(const float* __restrict__ h,
                          const float* __restrict__ att_src,
                          const float* __restrict__ att_dst,
                          float* __restrict__ as, float* __restrict__ ad,
                          float* __restrict__ m, float* __restrict__ denom,
                          int Nn, int H, int C) {
  const int n = blockIdx.x * blockDim.x + threadIdx.x;
  if (n >= Nn) return;
  const int HC = H * C;
  for (int hh = 0; hh < H; ++hh) {
    float s = 0.f, d = 0.f;
    for (int c = 0; c < C; ++c) {
      const float v = h[(size_t)n * HC + hh * C + c];
      s += v * att_src[hh * C + c];
      d += v * att_dst[hh * C + c];
    }
    as[n * H + hh] = s;
    ad[n * H + hh] = d;
    m[n * H + hh] = -INFINITY;
    denom[n * H + hh] = 0.f;
  }
}

// Edge pass 1: segment max of leaky_relu(a_s[src]+a_d[dst]) over dst.
// One thread per (edge, head); edges E..E+Nn-1 are the implicit self-loops.
__global__ void edge_max(const int* __restrict__ src, const int* __restrict__ dst,
                         const float* __restrict__ as, const float* __restrict__ ad,
                         float* __restrict__ m, int E, int Nn, int H) {
  const unsigned t = blockIdx.x * blockDim.x + threadIdx.x;
  const unsigned tot = (unsigned)(E + Nn) * (unsigned)H;
  if (t >= tot) return;
  const int hh = (int)(t % (unsigned)H);
  const int e  = (int)(t / (unsigned)H);
  int s, d;
  if (e < E) { s = src[e]; d = dst[e]; } else { s = d = e - E; }
  const float v = lrelu(as[s * H + hh] + ad[d * H + hh], ATT_SLOPE);
  atomicMaxF(&m[d * H + hh], v);
}

// Edge pass 2 (fused): w = exp(e - m[dst]); denom[dst] += w;
// acc[dst,:] += w * h[src,:].  One thread per (edge, feature): consecutive
// lanes of a wave cover consecutive features of the same edge, so both the
// h[src] gather and the atomic scatter into acc[dst] are wave-coalesced
// (one L2 atomic packet per wave instead of 16 serialized per-lane atomics).
// Normalization happens in the epilogue (algebraically identical to the
// reference's per-edge alpha since alpha = w / denom).
__global__ void edge_accum(const int* __restrict__ src, const int* __restrict__ dst,
                           const float* __restrict__ as, const float* __restrict__ ad,
                           const float* __restrict__ m, const float* __restrict__ hfeat,
                           float* __restrict__ denom, float* __restrict__ acc,
                           int E, int Nn, int H, int C) {
  const int HC = H * C;
  const unsigned t = blockIdx.x * blockDim.x + threadIdx.x;
  const unsigned tot = (unsigned)(E + Nn) * (unsigned)HC;
  if (t >= tot) return;
  const int j = (int)(t % (unsigned)HC);  // feature index within node
  const int e = (int)(t / (unsigned)HC);
  const int hh = j / C;
  int s, d;
  if (e < E) { s = src[e]; d = dst[e]; } else { s = d = e - E; }
  const float v = lrelu(as[s * H + hh] + ad[d * H + hh], ATT_SLOPE);
  const float w = expf(v - m[d * H + hh]);
  if ((j - hh * C) == 0) atomicAdd(&denom[d * H + hh], w);
  atomicAdd(&acc[(size_t)d * HC + j], w * hfeat[(size_t)s * HC + j]);
}

// Epilogue layer 1: feat = leaky( acc/denom + bias, 0.01 ), in place over acc.
__global__ void finalize_feat(float* __restrict__ acc, const float* __restrict__ denom,
                              const float* __restrict__ bias, int Nn, int H, int C) {
  const unsigned t = blockIdx.x * blockDim.x + threadIdx.x;
  const int HC = H * C;
  if (t >= (unsigned)Nn * (unsigned)HC) return;
  const int n = (int)(t / (unsigned)HC);
  const int j = (int)(t % (unsigned)HC);
  const float v = acc[(size_t)n * HC + j] / denom[n * H + j / C] + bias[j];
  acc[(size_t)n * HC + j] = lrelu(v, ACT_SLOPE);
}

// Epilogue layer 2 (H==1) fused with final linear: out[n] = leaky(acc/den + b2) @ Wo + bo.
__global__ void finalize_proj(const float* __restrict__ acc, const float* __restrict__ denom,
                              const float* __restrict__ bias, const float* __restrict__ Wo,
                              const float* __restrict__ bo, float* __restrict__ out,
                              int Nn, int C) {
  const int n = blockIdx.x * blockDim.x + threadIdx.x;
  if (n >= Nn) return;
  const float dn = denom[n];
  float s = 0.f;
  for (int c = 0; c < C; ++c) {
    const float f = lrelu(acc[(size_t)n * C + c] / dn + bias[c], ACT_SLOPE);
    s += f * Wo[c];
  }
  out[n] = s + bo[0];
}

extern "C" void kernel_launch(void* const* d_in, const int* in_sizes, int n_in,
                              void* d_out, int out_size, void* d_ws, size_t ws_size,
                              hipStream_t stream) {
  const float* x    = (const float*)d_in[0];
  const int*   ei   = (const int*)d_in[1];
  const float* W1   = (const float*)d_in[2];
  const float* aS1  = (const float*)d_in[3];
  const float* aD1  = (const float*)d_in[4];
  const float* b1   = (const float*)d_in[5];
  const float* W2   = (const float*)d_in[6];
  const float* aS2  = (const float*)d_in[7];
  const float* aD2  = (const float*)d_in[8];
  const float* b2   = (const float*)d_in[9];
  const float* Wo   = (const float*)d_in[10];
  const float* bo   = (const float*)d_in[11];
  float* out = (float*)d_out;

  const int IN_C = 64, HID = 16, HEADS = 4;
  const int HC1 = HEADS * HID;  // 64
  const int Nn = in_sizes[0] / IN_C;  // 50000
  const int E  = in_sizes[1] / 2;     // 1600000
  const int* src = ei;
  const int* dst = ei + E;

  // Workspace partition (f32), ~36 MB total.
  float* p = (float*)d_ws;
  float* h1   = p; p += (size_t)Nn * HC1;
  float* acc1 = p; p += (size_t)Nn * HC1;   // becomes layer-1 activations in place
  float* h2   = p; p += (size_t)Nn * HID;
  float* acc2 = p; p += (size_t)Nn * HID;
  float* as1  = p; p += (size_t)Nn * HEADS;
  float* ad1  = p; p += (size_t)Nn * HEADS;
  float* m1   = p; p += (size_t)Nn * HEADS;
  float* dn1  = p; p += (size_t)Nn * HEADS;
  float* as2  = p; p += (size_t)Nn;
  float* ad2  = p; p += (size_t)Nn;
  float* m2   = p; p += (size_t)Nn;
  float* dn2  = p; p += (size_t)Nn;

  const int TB = 256;

  // ---- Layer 1 ----
  {
    dim3 g((Nn + 15) / 16, HC1 / 16);
    wmma_gemm_f32<64, 64><<<g, 32, 0, stream>>>(x, W1, h1, Nn);
  }
  {
    const unsigned n = (unsigned)Nn * (unsigned)HC1;
    zero_buf<<<(n + TB - 1) / TB, TB, 0, stream>>>(acc1, n);
  }
  node_prep<<<(Nn + TB - 1) / TB, TB, 0, stream>>>(h1, aS1, aD1, as1, ad1, m1, dn1,
                                                   Nn, HEADS, HID);
  {
    const unsigned n = (unsigned)(E + Nn) * (unsigned)HEADS;
    edge_max<<<(n + TB - 1) / TB, TB, 0, stream>>>(src, dst, as1, ad1, m1,
                                                   E, Nn, HEADS);
  }
  {
    const unsigned n = (unsigned)(E + Nn) * (unsigned)HC1;
    edge_accum<<<(n + TB - 1) / TB, TB, 0, stream>>>(src, dst, as1, ad1, m1,
                                                     h1, dn1, acc1,
                                                     E, Nn, HEADS, HID);
  }
  {
    const unsigned n = (unsigned)Nn * (unsigned)HC1;
    finalize_feat<<<(n + TB - 1) / TB, TB, 0, stream>>>(acc1, dn1, b1, Nn, HEADS, HID);
  }

  // ---- Layer 2 ----
  {
    dim3 g((Nn + 15) / 16, HID / 16);
    wmma_gemm_f32<64, 16><<<g, 32, 0, stream>>>(acc1, W2, h2, Nn);
  }
  {
    const unsigned n = (unsigned)Nn * (unsigned)HID;
    zero_buf<<<(n + TB - 1) / TB, TB, 0, stream>>>(acc2, n);
  }
  node_prep<<<(Nn + TB - 1) / TB, TB, 0, stream>>>(h2, aS2, aD2, as2, ad2, m2, dn2,
                                                   Nn, 1, HID);
  {
    const unsigned n = (unsigned)(E + Nn);
    edge_max<<<(n + TB - 1) / TB, TB, 0, stream>>>(src, dst, as2, ad2, m2,
                                                   E, Nn, 1);
  }
  {
    const unsigned n = (unsigned)(E + Nn) * (unsigned)HID;
    edge_accum<<<(n + TB - 1) / TB, TB, 0, stream>>>(src, dst, as2, ad2, m2,
                                                     h2, dn2, acc2,
                                                     E, Nn, 1, HID);
  }
  finalize_proj<<<(Nn + TB - 1) / TB, TB, 0, stream>>>(acc2, dn2, b2, Wo, bo, out,
                                                       Nn, HID);
}